// DetectionLoss_4827543241462
// MI455X (gfx1250) — compile-verified
//
#include <hip/hip_runtime.h>
#include <stdint.h>

// Problem constants (match reference: B=16, A=65536, G=64)
#define BB 16
#define AA 65536
#define GG 64
#define IOU_TH 0.5f

typedef __attribute__((ext_vector_type(2))) float    v2f;
typedef __attribute__((ext_vector_type(8))) float    v8f;
typedef __attribute__((ext_vector_type(4))) uint32_t v4u;
typedef __attribute__((ext_vector_type(4))) int      v4i;
typedef __attribute__((ext_vector_type(8))) int      v8i;

// ---------------- math helpers ----------------

__device__ __forceinline__ float diou_loss(float4 p, float4 g) {
    float pa = (p.z - p.x) * (p.w - p.y);
    float ga = (g.z - g.x) * (g.w - g.y);
    float lx = fmaxf(p.x, g.x), ly = fmaxf(p.y, g.y);
    float rx = fminf(p.z, g.z), ry = fminf(p.w, g.w);
    float iw = fmaxf(rx - lx, 0.f), ih = fmaxf(ry - ly, 0.f);
    float inter = iw * ih;
    float iou = inter / (pa + ga - inter + 1e-6f);
    float dx = (p.x + p.z - g.x - g.z) * 0.5f;
    float dy = (p.y + p.w - g.y - g.w) * 0.5f;
    float cd = dx * dx + dy * dy;
    float ew = fmaxf(p.z, g.z) - fminf(p.x, g.x);
    float eh = fmaxf(p.w, g.w) - fminf(p.y, g.y);
    return 1.f - iou + cd / (ew * ew + eh * eh + 1e-6f);
}

// focal with targets==1; cp -> logit -> alpha*ce*(1-p)^2
__device__ __forceinline__ float focal_pos(float cp) {
    float logit = logf(cp / (1.f - cp + 1e-10f));
    float p = 1.f / (1.f + expf(-logit));
    float ls = (logit >= 0.f) ? -log1pf(expf(-logit)) : logit - log1pf(expf(logit));
    float q = 1.f - p;
    return 0.25f * (-ls) * q * q;
}
// focal with targets==0
__device__ __forceinline__ float focal_neg(float cp) {
    float logit = logf(cp / (1.f - cp + 1e-10f));
    float p = 1.f / (1.f + expf(-logit));
    float lsn = (logit <= 0.f) ? -log1pf(expf(logit)) : -logit - log1pf(expf(-logit));
    return 0.75f * (-lsn) * p * p;
}

__device__ __forceinline__ unsigned long long shfl_xor_u64(unsigned long long v, int m) {
    unsigned lo = (unsigned)v, hi = (unsigned)(v >> 32);
    lo = __shfl_xor(lo, m);
    hi = __shfl_xor(hi, m);
    return ((unsigned long long)hi << 32) | lo;
}

// ---------------- WMMA block reduction (256 fp32 -> 1, exact fp32) ----------------
// Stage partials in LDS as 4 tiles of 16x4; wave 0 chains 4x V_WMMA_F32_16X16X4_F32
// with B = ones so D[i][*] = row-sums; one xor-16 lane add finishes the 256-sum.
__device__ __forceinline__ float block_sum_wmma(float v, float* sm) {
    sm[threadIdx.x] = v;
    __syncthreads();
    float total = 0.f;
    if (threadIdx.x < 32u) {
#if __has_builtin(__builtin_amdgcn_wmma_f32_16x16x4_f32)
        const v2f ones = {1.f, 1.f};
        v8f c = {};
        int m  = threadIdx.x & 15;          // A-matrix row (lanes 0-15 / 16-31)
        int kb = (threadIdx.x >> 4) << 1;   // K pair: lanes 0-15 -> K0,K1; 16-31 -> K2,K3
#pragma unroll
        for (int t = 0; t < 4; ++t) {
            v2f a;
            a.x = sm[t * 64 + m * 4 + kb];
            a.y = sm[t * 64 + m * 4 + kb + 1];
            c = __builtin_amdgcn_wmma_f32_16x16x4_f32(false, a, false, ones,
                                                      (short)0, c, false, false);
        }
        float s = c[0] + c[1] + c[2] + c[3] + c[4] + c[5] + c[6] + c[7];
        s += __shfl_xor(s, 16);  // rows 0-7 half + rows 8-15 half
        total = s;
#else
        float s = 0.f;
        for (int i = threadIdx.x; i < 256; i += 32) s += sm[i];
        for (int o = 16; o; o >>= 1) s += __shfl_xor(s, o);
        total = s;
#endif
    }
    __syncthreads();
    return total;  // valid on wave 0
}

// ---------------- TDM: stage 64 GT boxes (256 floats / 1KB) into LDS ----------------
__device__ __forceinline__ void tdm_load_gt(const float* gsrc, float* ldst) {
#if __has_builtin(__builtin_amdgcn_tensor_load_to_lds)
    if (threadIdx.x == 0) {
        uint64_t ga = (uint64_t)gsrc;
        uint32_t lo = (uint32_t)(uint64_t)(uintptr_t)ldst;  // LDS offset = addr[31:0]
        v4u g0;
        g0.x = 1u;                 // count=1, user descriptor
        g0.y = lo;                 // lds_addr [63:32]
        g0.z = (uint32_t)ga;       // global_addr [95:64]
        g0.w = (uint32_t)((ga >> 32) & 0x1FFFFFFull) | (2u << 30);  // addr[56:32] | type=2
        v8i g1;
        g1[0] = 0x00020000;        // wg_mask=0, data_size=2 (4B)
        g1[1] = (int)(256u << 16); // tensor_dim0 = 256 (bits 63:48)
        g1[2] = 0x00010000;        // tensor_dim0 hi=0 | tensor_dim1 = 1
        g1[3] = (int)(256u << 16); // tensor_dim1 hi=0 | tile_dim0 = 256
        g1[4] = 0;                 // tile_dim1=0, tile_dim2=0 (1-D tile)
        g1[5] = 256;               // tensor_dim0_stride = 256
        g1[6] = 0; g1[7] = 0;
        v4i z4 = {0, 0, 0, 0};
#if defined(__clang_major__) && (__clang_major__ >= 23)
        v8i z8 = {0, 0, 0, 0, 0, 0, 0, 0};
        __builtin_amdgcn_tensor_load_to_lds(g0, g1, z4, z4, z8, 0);
#else
        __builtin_amdgcn_tensor_load_to_lds(g0, g1, z4, z4, 0);
#endif
        __builtin_amdgcn_s_wait_tensorcnt(0);
    }
#else
    if (threadIdx.x < 256) ldst[threadIdx.x] = gsrc[threadIdx.x];
#endif
    __syncthreads();
}

// ---------------- kernel 1: per-anchor best IoU / argmax over 64 GT ----------------
__global__ __launch_bounds__(256) void k_best_per_anchor(
    const float* __restrict__ anchors, const float* __restrict__ gt,
    float* __restrict__ biou, int* __restrict__ bidx) {
    __shared__ float sgt[GG * 4];
    __shared__ float sga[GG];
    int b = blockIdx.y;
    tdm_load_gt(gt + (size_t)b * GG * 4, sgt);
    if (threadIdx.x < GG) {
        float x0 = sgt[threadIdx.x * 4 + 0], y0 = sgt[threadIdx.x * 4 + 1];
        float x1 = sgt[threadIdx.x * 4 + 2], y1 = sgt[threadIdx.x * 4 + 3];
        sga[threadIdx.x] = (x1 - x0) * (y1 - y0);
    }
    __syncthreads();
    int a = blockIdx.x * 256 + threadIdx.x;
    float4 ab = ((const float4*)anchors)[a];
    float aarea = (ab.z - ab.x) * (ab.w - ab.y);
    float best = -1.f;
    int bg = 0;
#pragma unroll 4
    for (int g = 0; g < GG; ++g) {
        float gx0 = sgt[g * 4 + 0], gy0 = sgt[g * 4 + 1];
        float gx1 = sgt[g * 4 + 2], gy1 = sgt[g * 4 + 3];
        float lx = fmaxf(ab.x, gx0), ly = fmaxf(ab.y, gy0);
        float rx = fminf(ab.z, gx1), ry = fminf(ab.w, gy1);
        float iw = fmaxf(rx - lx, 0.f), ih = fmaxf(ry - ly, 0.f);
        float inter = iw * ih;
        float v = inter / (aarea + sga[g] - inter + 1e-6f);
        if (v > best) { best = v; bg = g; }   // strict > keeps first occurrence
    }
    size_t i = (size_t)b * AA + a;
    biou[i] = best;
    bidx[i] = bg;
}

// ---------------- kernel 2: best anchor per GT -> force-match (scatter pos) --------
__global__ __launch_bounds__(256) void k_best_anchor_per_gt(
    const float* __restrict__ anchors, const float* __restrict__ gt,
    unsigned char* __restrict__ pos) {
    __shared__ unsigned long long wk[8];
    int g = blockIdx.x, b = blockIdx.y;
    const float* gp = gt + ((size_t)b * GG + g) * 4;
    float gx0 = gp[0], gy0 = gp[1], gx1 = gp[2], gy1 = gp[3];
    float garea = (gx1 - gx0) * (gy1 - gy0);
    unsigned long long key = 0ull;
    for (int a = threadIdx.x; a < AA; a += 256) {
        float4 ab = ((const float4*)anchors)[a];
        float aarea = (ab.z - ab.x) * (ab.w - ab.y);
        float lx = fmaxf(ab.x, gx0), ly = fmaxf(ab.y, gy0);
        float rx = fminf(ab.z, gx1), ry = fminf(ab.w, gy1);
        float iw = fmaxf(rx - lx, 0.f), ih = fmaxf(ry - ly, 0.f);
        float inter = iw * ih;
        float v = inter / (aarea + garea - inter + 1e-6f);
        // pack (iou_bits, ~idx): max -> largest iou, smallest idx on tie (first-occurrence)
        unsigned long long k =
            ((unsigned long long)__float_as_uint(v) << 32) | (unsigned)(~a);
        if (k > key) key = k;
    }
    for (int o = 16; o; o >>= 1) {
        unsigned long long other = shfl_xor_u64(key, o);
        if (other > key) key = other;
    }
    if ((threadIdx.x & 31) == 0) wk[threadIdx.x >> 5] = key;
    __syncthreads();
    if (threadIdx.x == 0) {
        for (int w = 1; w < 8; ++w)
            if (wk[w] > key) key = wk[w];
        unsigned besta = ~(unsigned)(key & 0xFFFFFFFFull);
        pos[(size_t)b * AA + besta] = 1;
    }
}

// ---------------- kernel 3: positive pass (DIoU + pos-focal + n_pos) ---------------
__global__ __launch_bounds__(256) void k_pos_loss(
    const float* __restrict__ bbox, const float* __restrict__ conf,
    const float* __restrict__ gt, const float* __restrict__ biou,
    const int* __restrict__ bidx, unsigned char* __restrict__ pos,
    float* __restrict__ loc_acc, float* __restrict__ posf_acc,
    float* __restrict__ npos_acc) {
    __shared__ float sm[256];
    int b = blockIdx.y;
    int a = blockIdx.x * 256 + threadIdx.x;
    size_t i = (size_t)b * AA + a;
    unsigned char p = pos[i] | (biou[i] > IOU_TH ? 1 : 0);
    pos[i] = p;
    float locv = 0.f, pf = 0.f, npv = 0.f;
    if (p) {
        int gi = bidx[i];
        float4 gb = ((const float4*)gt)[(size_t)b * GG + gi];
        float4 pb = ((const float4*)bbox)[i];
        locv = diou_loss(pb, gb);
        pf = focal_pos(conf[i]);
        npv = 1.f;
    }
    float t;
    t = block_sum_wmma(locv, sm); if (threadIdx.x == 0) atomicAdd(&loc_acc[b], t);
    t = block_sum_wmma(pf, sm);   if (threadIdx.x == 0) atomicAdd(&posf_acc[b], t);
    t = block_sum_wmma(npv, sm);  if (threadIdx.x == 0) atomicAdd(&npos_acc[b], t);
}

// ---------------- radix-select (exact top-num_neg by conf bit pattern) -------------
__global__ void k_init_select(const float* __restrict__ npos_acc,
                              int* __restrict__ selk, unsigned* __restrict__ prefix) {
    int b = threadIdx.x;
    if (b >= BB) return;
    int np = (int)(npos_acc[b] + 0.5f);
    int k = 3 * np;
    int lim = AA - np;
    if (k > lim) k = lim;
    if (k < 0) k = 0;
    selk[b] = k;
    prefix[b] = 0u;
}

__global__ __launch_bounds__(256) void k_hist(
    const float* __restrict__ conf, const unsigned char* __restrict__ pos,
    const unsigned* __restrict__ prefix, unsigned* __restrict__ hist, int p) {
    __shared__ unsigned lh[256];
    lh[threadIdx.x] = 0u;
    __syncthreads();
    int b = blockIdx.y;
    unsigned pref = prefix[b];
    int shift = 24 - 8 * p;
    for (int a = blockIdx.x * 256 + threadIdx.x; a < AA; a += 16 * 256) {
        size_t i = (size_t)b * AA + a;
        if (!pos[i]) {
            unsigned bits = __float_as_uint(conf[i]);  // cp>0 -> uint order == float order
            if (p == 0 || (bits >> (shift + 8)) == (pref >> (shift + 8)))
                atomicAdd(&lh[(bits >> shift) & 255u], 1u);
        }
    }
    __syncthreads();
    unsigned c = lh[threadIdx.x];
    if (c) atomicAdd(&hist[((size_t)p * BB + b) * 256 + threadIdx.x], c);
}

__global__ void k_scan(const unsigned* __restrict__ hist, unsigned* __restrict__ prefix,
                       int* __restrict__ selk, int p) {
    int b = threadIdx.x;
    if (b >= BB) return;
    const unsigned* h = hist + ((size_t)p * BB + b) * 256;
    int k = selk[b];
    if (k <= 0) return;
    int cum = 0, chosen = 0;
    for (int v = 255; v >= 0; --v) {
        int c = (int)h[v];
        if (cum + c >= k) { chosen = v; k -= cum; break; }
        cum += c;
    }
    prefix[b] |= (unsigned)chosen << (24 - 8 * p);
    selk[b] = k;  // remaining count to take at value == T
}

// ---------------- kernel 5: negative focal sum over bits > T -----------------------
__global__ __launch_bounds__(256) void k_neg_loss(
    const float* __restrict__ conf, const unsigned char* __restrict__ pos,
    const unsigned* __restrict__ prefix, float* __restrict__ negf_acc) {
    __shared__ float sm[256];
    int b = blockIdx.y;
    int a = blockIdx.x * 256 + threadIdx.x;
    size_t i = (size_t)b * AA + a;
    float nf = 0.f;
    if (!pos[i]) {
        unsigned bits = __float_as_uint(conf[i]);
        if (bits > prefix[b]) nf = focal_neg(conf[i]);
    }
    float t = block_sum_wmma(nf, sm);
    if (threadIdx.x == 0) atomicAdd(&negf_acc[b], t);
}

// ---------------- finalize ---------------------------------------------------------
__global__ void k_final(const float* __restrict__ loc, const float* __restrict__ posf,
                        const float* __restrict__ npos_acc, const float* __restrict__ negf,
                        const int* __restrict__ selk, const unsigned* __restrict__ prefix,
                        float* __restrict__ out) {
    int b = threadIdx.x;
    float l = 0.f, cf = 0.f, np = 0.f;
    if (b < BB) {
        l = loc[b];
        float boundary = 0.f;
        if (selk[b] > 0)  // ties at the k-th value contribute identical focal values
            boundary = (float)selk[b] * focal_neg(__uint_as_float(prefix[b]));
        cf = posf[b] + negf[b] + boundary;
        np = npos_acc[b];
    }
    for (int o = 16; o; o >>= 1) {
        l += __shfl_xor(l, o);
        cf += __shfl_xor(cf, o);
        np += __shfl_xor(np, o);
    }
    if (threadIdx.x == 0) {
        float npf = fmaxf(1.f, np);
        out[0] = 1.5f * l / npf + cf / npf;
    }
}

// ---------------- host launch ------------------------------------------------------
extern "C" void kernel_launch(void* const* d_in, const int* in_sizes, int n_in,
                              void* d_out, int out_size, void* d_ws, size_t ws_size,
                              hipStream_t stream) {
    (void)in_sizes; (void)n_in; (void)out_size; (void)ws_size;
    const float* bbox    = (const float*)d_in[0];  // [B,A,4]
    const float* conf    = (const float*)d_in[1];  // [B,A]
    const float* anchors = (const float*)d_in[2];  // [A,4]
    const float* gt      = (const float*)d_in[3];  // [B,G,4]
    float* out = (float*)d_out;

    // workspace layout (bytes)
    const size_t OFF_POS  = 0;                       // B*A u8 pos/forced flags
    const size_t OFF_HIST = OFF_POS + (size_t)BB * AA;          // 4*B*256 u32
    const size_t OFF_ACC  = OFF_HIST + 4ull * BB * 256 * 4;     // accumulators
    const size_t ZERO_SZ  = OFF_ACC + 384;                      // zeroed region
    const size_t OFF_BIOU = ZERO_SZ;                            // B*A f32
    const size_t OFF_BIDX = OFF_BIOU + (size_t)BB * AA * 4;     // B*A i32
    // total ~9.1 MB

    char* ws = (char*)d_ws;
    unsigned char* pos = (unsigned char*)(ws + OFF_POS);
    unsigned* hist     = (unsigned*)(ws + OFF_HIST);
    float* loc_acc     = (float*)(ws + OFF_ACC);
    float* posf_acc    = loc_acc + 16;
    float* npos_acc    = loc_acc + 32;
    float* negf_acc    = loc_acc + 48;
    int* selk          = (int*)(loc_acc + 64);
    unsigned* prefix   = (unsigned*)(loc_acc + 80);
    float* biou        = (float*)(ws + OFF_BIOU);
    int* bidx          = (int*)(ws + OFF_BIDX);

    hipMemsetAsync(d_ws, 0, ZERO_SZ, stream);

    dim3 gridA(AA / 256, BB);
    k_best_per_anchor<<<gridA, 256, 0, stream>>>(anchors, gt, biou, bidx);
    k_best_anchor_per_gt<<<dim3(GG, BB), 256, 0, stream>>>(anchors, gt, pos);
    k_pos_loss<<<gridA, 256, 0, stream>>>(bbox, conf, gt, biou, bidx, pos,
                                          loc_acc, posf_acc, npos_acc);
    k_init_select<<<1, 16, 0, stream>>>(npos_acc, selk, prefix);
    for (int p = 0; p < 4; ++p) {
        k_hist<<<dim3(16, BB), 256, 0, stream>>>(conf, pos, prefix, hist, p);
        k_scan<<<1, 16, 0, stream>>>(hist, prefix, selk, p);
    }
    k_neg_loss<<<gridA, 256, 0, stream>>>(conf, pos, prefix, negf_acc);
    k_final<<<1, 32, 0, stream>>>(loc_acc, posf_acc, npos_acc, negf_acc,
                                  selk, prefix, out);
}